// _UpFirDnNd_47356309405977
// MI455X (gfx1250) — compile-verified
//
#include <hip/hip_runtime.h>
#include <cstdint>

// upfirdn2d: up=(2,2), down=(1,1), pad=(2,1,2,1), gain=1, kernel 4x4 (flipped)
// x: (4,64,256,256) f32  ->  out: (4,64,512,512) f32
//
// Bandwidth-bound (>=320MB traffic, ~0.54 GFLOP). Strategy:
//  - per-phase 2x2-tap decomposition (4 FMA / output pixel, zero wasted MACs)
//  - async global->LDS staging of the input tile (CDNA5 ASYNCcnt path)
//  - non-temporal B128 output stores (keep 64MB input resident in 192MB L2)

typedef float v4f __attribute__((ext_vector_type(4)));

#define B_   4
#define C_   64
#define H_   256
#define W_   256
#define OH_  512
#define OW_  512

#define OUT_TW 128     // output tile width  (per block)
#define OUT_TH 32      // output tile height (per block)
#define IN_TW  66      // = OUT_TW/2 + 2 (halo)
#define IN_TH  18      // = OUT_TH/2 + 2 (halo)
#define LDS_W  68      // padded LDS row stride (floats)

__global__ __launch_bounds__(256) void upfirdn2d_up2_k4(
    const float* __restrict__ x, const float* __restrict__ kern,
    float* __restrict__ out) {
  __shared__ float tile[IN_TH * LDS_W];   // 4896 B

  const int tid = threadIdx.x;            // 0..255
  const int img = blockIdx.z;             // 0..255  (b*C + c)
  const int OX0 = blockIdx.x * OUT_TW;    // output tile origin
  const int OY0 = blockIdx.y * OUT_TH;
  const int IX0 = (OX0 >> 1) - 1;         // staged input origin (incl. halo)
  const int IY0 = (OY0 >> 1) - 1;

  // Flipped + gain-scaled 4x4 weights: kk[i][j] = 4 * k[3-i][3-j]
  float wk[4][4];
#pragma unroll
  for (int i = 0; i < 4; ++i)
#pragma unroll
    for (int j = 0; j < 4; ++j)
      wk[i][j] = 4.0f * kern[(3 - i) * 4 + (3 - j)];

  // ---- Stage 66x18 input tile into LDS via async global->LDS loads ----
  const uint32_t lds_base = (uint32_t)(uintptr_t)&tile[0];
  const uint64_t sbase =
      (uint64_t)(uintptr_t)(x + (size_t)img * (H_ * W_));

  for (int e = tid; e < IN_TH * IN_TW; e += 256) {
    const int r = e / IN_TW;
    const int c = e - r * IN_TW;
    const int iy = IY0 + r;
    const int ix = IX0 + c;
    const uint32_t laddr = lds_base + (uint32_t)(r * LDS_W + c) * 4u;
    if ((unsigned)iy < (unsigned)H_ && (unsigned)ix < (unsigned)W_) {
      const uint32_t voff = (uint32_t)((iy * W_ + ix) * 4);
      // GVS mode: mem = SGPR64 base + VGPR32 offset ; LDS dest addr in vdst VGPR.
      asm volatile("global_load_async_to_lds_b32 %0, %1, %2"
                   :: "v"(laddr), "v"(voff), "s"(sbase)
                   : "memory");
    } else {
      tile[r * LDS_W + c] = 0.0f;   // zero-pad halo (DS store)
    }
  }
  asm volatile("s_wait_asynccnt 0" ::: "memory");
  __syncthreads();

  // ---- Compute: each thread -> 4 output rows x float4 of columns ----
  const int tx = tid & 31;   // 32 threads across -> 128 cols (float4 each)
  const int ty = tid >> 5;   // 8 row-groups -> 32 rows (4 each)
  const int lx = tx * 2;     // staged col of (X-1) for ox_local = 4*tx

  float* outp = out + ((size_t)img * OH_ + OY0) * OW_ + OX0 + tx * 4;

#pragma unroll
  for (int rr = 0; rr < 4; ++rr) {
    const int oyl = ty * 4 + rr;            // output row within tile
    const int p   = rr & 1;                 // vertical phase (OY0 even)
    const int li  = ty * 2 + ((rr + 1) >> 1);  // staged row of iy0
    const float* r0 = &tile[li * LDS_W + lx];
    const float* r1 = r0 + LDS_W;

    const float c0 = r0[0], c1 = r0[1], c2 = r0[2], c3 = r0[3];
    const float d0 = r1[0], d1 = r1[1], d2 = r1[2], d3 = r1[3];

    const float a0 = wk[p][0], a1 = wk[p][1], a2 = wk[p][2], a3 = wk[p][3];
    const float b0 = wk[p + 2][0], b1 = wk[p + 2][1];
    const float b2 = wk[p + 2][2], b3 = wk[p + 2][3];

    v4f o;
    o.x = c0 * a0 + c1 * a2 + d0 * b0 + d1 * b2;   // even col phase
    o.y = c1 * a1 + c2 * a3 + d1 * b1 + d2 * b3;   // odd col phase
    o.z = c1 * a0 + c2 * a2 + d1 * b0 + d2 * b2;
    o.w = c2 * a1 + c3 * a3 + d2 * b1 + d3 * b3;

    // Streaming store: output never re-read; keep input resident in L2.
    __builtin_nontemporal_store(o, (v4f*)(outp + (size_t)oyl * OW_));
  }
}

extern "C" void kernel_launch(void* const* d_in, const int* in_sizes, int n_in,
                              void* d_out, int out_size, void* d_ws, size_t ws_size,
                              hipStream_t stream) {
  const float* x    = (const float*)d_in[0];   // (4,64,256,256)
  const float* kern = (const float*)d_in[1];   // (4,4)
  float* out        = (float*)d_out;           // (4,64,512,512)

  dim3 grid(OW_ / OUT_TW, OH_ / OUT_TH, B_ * C_);  // (4, 16, 256)
  upfirdn2d_up2_k4<<<grid, dim3(256, 1, 1), 0, stream>>>(x, kern, out);
}